// TeacherMLP_44813688767073
// MI455X (gfx1250) — compile-verified
//
#include <hip/hip_runtime.h>
#include <math.h>

// ---------------------------------------------------------------------------
// TeacherMLP on MI455X (gfx1250, wave32)
//
//  * W_eff precomputed (constants) -> 3 tiny elementwise kernels
//  * hidden layers: sum_i tanh(h*We) -> trans-pipe bound, uses V_TANH_F32
//  * output layer: true f32 GEMM     -> V_WMMA_F32_16X16X4_F32
// ---------------------------------------------------------------------------

typedef float v2f __attribute__((ext_vector_type(2)));
typedef float v8f __attribute__((ext_vector_type(8)));

#define BATCH   2048
#define D_IN    256
#define D_HID   256
#define D_OUT   128

__device__ __forceinline__ float fast_tanh(float x) {
#if __has_builtin(__builtin_amdgcn_tanhf)
  return __builtin_amdgcn_tanhf(x);
#elif __has_builtin(__builtin_amdgcn_tanh_f32)
  return __builtin_amdgcn_tanh_f32(x);
#else
  float r;
  // gfx1250 hardware tanh; v_nop covers the trans-result-use hazard
  asm volatile("v_tanh_f32 %0, %1\n\tv_nop" : "=v"(r) : "v"(x));
  return r;
#endif
}

// ---------------------------------------------------------------------------
// W_eff[o,i] = [W, sin(W), tanh(W), W*W][idx[o,i]]   (one-time, elementwise)
// ---------------------------------------------------------------------------
__global__ __launch_bounds__(256) void weff_kernel(const float* __restrict__ W,
                                                   const int* __restrict__ idx,
                                                   float* __restrict__ We, int n) {
  int t = blockIdx.x * 256 + threadIdx.x;
  if (t >= n) return;
  float w = W[t];
  int k = idx[t];
  float r;
  if (k == 0)      r = w;
  else if (k == 1) r = sinf(w);
  else if (k == 2) r = tanhf(w);
  else             r = w * w;
  We[t] = r;
}

// ---------------------------------------------------------------------------
// Hidden layer: h_out[b,o] = bias[o] + sum_i tanh(h_in[b,i] * We[o,i])
// Block = 256 threads, tile = 64 batch x 32 out, K chunked by 128.
// LDS: sh_h[64][128] (wave-uniform broadcast reads)  = 32 KB
//      sh_w[32][129] (lane-varying, padded, 0-confl) = 16.1 KB
// Each thread: 32 outputs? no -> one out column, 8 batch rows => 8 accums
// of ILP to hide trans latency. 2048 tanh per thread.
// ---------------------------------------------------------------------------
#define TB 64
#define TO 32
#define KC 128

__global__ __launch_bounds__(256) void hidden_layer_kernel(
    const float* __restrict__ h_in,   // [BATCH, 256]
    const float* __restrict__ We,     // [256, 256]
    const float* __restrict__ bias,   // [256]
    float* __restrict__ h_out)        // [BATCH, 256]
{
  __shared__ float sh_h[TB][KC];
  __shared__ float sh_w[TO][KC + 1];

  const int tid = threadIdx.x;
  const int b0  = blockIdx.x * TB;
  const int o0  = blockIdx.y * TO;
  const int ol  = tid & 31;   // output lane (varies within wave)
  const int bg  = tid >> 5;   // batch group 0..7 (uniform within wave)

  float acc[8];
#pragma unroll
  for (int j = 0; j < 8; ++j) acc[j] = 0.0f;

  for (int kc = 0; kc < D_IN; kc += KC) {
    __syncthreads();
    // cooperative loads (coalesced: consecutive tid -> consecutive i)
    for (int t = tid; t < TB * KC; t += 256) {
      int b = t >> 7, i = t & (KC - 1);
      sh_h[b][i] = h_in[(b0 + b) * D_IN + kc + i];
    }
    for (int t = tid; t < TO * KC; t += 256) {
      int o = t >> 7, i = t & (KC - 1);
      sh_w[o][i] = We[(o0 + o) * D_IN + kc + i];
    }
    __syncthreads();

    for (int i = 0; i < KC; ++i) {
      float wv = sh_w[ol][i];          // conflict-free (stride 129)
#pragma unroll
      for (int j = 0; j < 8; ++j) {    // sh_h reads are wave-uniform broadcasts
        acc[j] += fast_tanh(sh_h[bg * 8 + j][i] * wv);
      }
    }
  }

  float bv = bias[o0 + ol];
#pragma unroll
  for (int j = 0; j < 8; ++j) {
    int b = b0 + bg * 8 + j;
    h_out[b * D_HID + o0 + ol] = acc[j] + bv;   // coalesced store
  }
}

// ---------------------------------------------------------------------------
// Output layer: out = H(2048x256) * We2^T(256x128) + b2, pure f32 GEMM.
// One wave per 16(M) x 32(N) tile pair: V_WMMA_F32_16X16X4_F32, K-loop of 64.
// A frag: lane<16 holds K={0,1}, lane>=16 holds K={2,3} of row M=lane%16.
// B frag: same pattern with column N=lane%16 taken from We2 rows (B = We2^T).
// C/D:    vgpr r -> M = r + 8*(lane>>4), N = lane&15.
// ---------------------------------------------------------------------------
__global__ __launch_bounds__(256) void out_layer_wmma_kernel(
    const float* __restrict__ h,      // [BATCH, 256]
    const float* __restrict__ We2,    // [128, 256]
    const float* __restrict__ bias,   // [128]
    float* __restrict__ out)          // [BATCH, 128]
{
  const int lane = threadIdx.x & 31;
  const int gw   = blockIdx.x * 8 + (threadIdx.x >> 5);  // global wave id
  const int tm   = gw >> 2;          // 0..127  (M tile, 16 rows)
  const int tnp  = gw & 3;           // 0..3    (N tile pair, 32 cols)

  const int mrow  = lane & 15;
  const int khalf = lane >> 4;       // 0 or 1 -> K offset 0 or 2

  const float* arow  = h   + (tm * 16 + mrow) * D_HID;
  const float* brow0 = We2 + (tnp * 32 + mrow) * D_HID;
  const float* brow1 = brow0 + 16 * D_HID;

  v8f c0 = {};
  v8f c1 = {};

#pragma unroll 8
  for (int k = 0; k < D_HID; k += 4) {
    v2f a  = *(const v2f*)(arow  + k + 2 * khalf);
    v2f bA = *(const v2f*)(brow0 + k + 2 * khalf);
    v2f bB = *(const v2f*)(brow1 + k + 2 * khalf);
    c0 = __builtin_amdgcn_wmma_f32_16x16x4_f32(false, a, false, bA,
                                               (short)0, c0, false, false);
    c1 = __builtin_amdgcn_wmma_f32_16x16x4_f32(false, a, false, bB,
                                               (short)0, c1, false, false);
  }

  const int col0 = tnp * 32 + (lane & 15);
  const float bv0 = bias[col0];
  const float bv1 = bias[col0 + 16];
#pragma unroll
  for (int r = 0; r < 8; ++r) {
    int row = tm * 16 + r + 8 * khalf;
    out[row * D_OUT + col0]      = c0[r] + bv0;
    out[row * D_OUT + col0 + 16] = c1[r] + bv1;
  }
}

// ---------------------------------------------------------------------------
// launcher
// inputs (setup_inputs dict order):
//  0:x  1:W0 2:b0 3:f_idx0  4:W1 5:b1 6:f_idx1  7:W2 8:b2 9:f_idx2
// ---------------------------------------------------------------------------
extern "C" void kernel_launch(void* const* d_in, const int* in_sizes, int n_in,
                              void* d_out, int out_size, void* d_ws, size_t ws_size,
                              hipStream_t stream) {
  const float* x  = (const float*)d_in[0];
  const float* W0 = (const float*)d_in[1];
  const float* b0 = (const float*)d_in[2];
  const int*   i0 = (const int*)  d_in[3];
  const float* W1 = (const float*)d_in[4];
  const float* b1 = (const float*)d_in[5];
  const int*   i1 = (const int*)  d_in[6];
  const float* W2 = (const float*)d_in[7];
  const float* b2 = (const float*)d_in[8];
  const int*   i2 = (const int*)  d_in[9];

  char* ws = (char*)d_ws;
  float* We0 = (float*)ws;  ws += (size_t)D_HID * D_IN  * sizeof(float);
  float* We1 = (float*)ws;  ws += (size_t)D_HID * D_HID * sizeof(float);
  float* We2 = (float*)ws;  ws += (size_t)D_OUT * D_HID * sizeof(float);
  float* h1  = (float*)ws;  ws += (size_t)BATCH * D_HID * sizeof(float);
  float* h2  = (float*)ws;

  // fold the atom nonlinearity into effective weights (constants)
  weff_kernel<<<(D_HID * D_IN  + 255) / 256, 256, 0, stream>>>(W0, i0, We0, D_HID * D_IN);
  weff_kernel<<<(D_HID * D_HID + 255) / 256, 256, 0, stream>>>(W1, i1, We1, D_HID * D_HID);
  weff_kernel<<<(D_OUT * D_HID + 255) / 256, 256, 0, stream>>>(W2, i2, We2, D_OUT * D_HID);

  dim3 grid(BATCH / TB, D_HID / TO);  // 32 x 8
  hidden_layer_kernel<<<grid, 256, 0, stream>>>(x,  We0, b0, h1);
  hidden_layer_kernel<<<grid, 256, 0, stream>>>(h1, We1, b1, h2);

  // 512 waves, 8 waves/block -> 64 blocks
  out_layer_wmma_kernel<<<64, 256, 0, stream>>>(h2, We2, b2, (float*)d_out);
}